// SimpleAttractorCore_14010183320234
// MI455X (gfx1250) — compile-verified
//
#include <hip/hip_runtime.h>

// CDNA5 / gfx1250, wave32. One wave integrates a tile of 16 points (D=128)
// for all iterations, using V_WMMA_F32_16X16X4_F32 for both GEMMs per step.

typedef float v2f __attribute__((ext_vector_type(2)));
typedef float v8f __attribute__((ext_vector_type(8)));

#define GRAVITY 0.001f
#define STEP    0.01f
#define EPS     1e-8f

#define NA 10          // real attractors (padded to 16)
#define DIM 128
#define KT  (DIM / 4)  // 32 k-tiles for GEMM1 (K=4 per WMMA)
#define DT  (DIM / 16) // 8 dim-tiles for GEMM2

__global__ __launch_bounds__(128)
void attractor_wmma_kernel(const float* __restrict__ emb,
                           const float* __restrict__ attr,
                           const int*   __restrict__ n_iter_p,
                           float*       __restrict__ out)
{
    // wave-private LDS tiles (4 waves / block)
    __shared__ float W_lds[4][16][16];    // w[s][a]
    __shared__ float F_lds[4][16][DIM];   // (W@At)[s][d]

    const int lane = threadIdx.x & 31;
    const int wave = threadIdx.x >> 5;
    const int s    = lane & 15;   // point-in-tile index / N index
    const int h    = lane >> 4;   // fragment half (K/M high half select)
    const long base = ((long)blockIdx.x * 4 + wave) * 16;  // first point of tile

    const int n_iter = *n_iter_p;

    // ---- positions as A-fragments: pa[kt] = P[base+s][4kt+2h .. +1] ----
    v2f pa[KT];
#pragma unroll
    for (int kt = 0; kt < KT; ++kt)
        pa[kt] = *(const v2f*)(emb + (base + s) * DIM + 4 * kt + 2 * h);

    // ---- attractors as B-fragments for GEMM1: b1[kt] = At[a=s][4kt+2h..+1] ----
    v2f b1[KT];
    float an_part = 0.f;
#pragma unroll
    for (int kt = 0; kt < KT; ++kt) {
        v2f v = {0.f, 0.f};
        if (s < NA) v = *(const v2f*)(attr + s * DIM + 4 * kt + 2 * h);
        b1[kt] = v;
        an_part += v.x * v.x + v.y * v.y;
    }
    // |At[a]|^2 for a = s (fragment halves hold complementary dims)
    const float anorm2 = an_part + __shfl_xor(an_part, 16, 32);

    // ---- attractors^T as A-fragments for GEMM2: a2[dt][kt] = At[a0..a0+1][dt*16+s] ----
    v2f a2[DT][4];
#pragma unroll
    for (int dt = 0; dt < DT; ++dt) {
#pragma unroll
        for (int kt = 0; kt < 4; ++kt) {
            const int a0 = 4 * kt + 2 * h;
            const int d  = dt * 16 + s;
            v2f v;
            v.x = (a0     < NA) ? attr[(a0    ) * DIM + d] : 0.f;
            v.y = (a0 + 1 < NA) ? attr[(a0 + 1) * DIM + d] : 0.f;
            a2[dt][kt] = v;
        }
    }

    const float amask = (s < NA) ? 1.f : 0.f;  // kill padded-attractor forces

    for (int it = 0; it < n_iter; ++it) {
        // ---- |p|^2 per point (lane halves hold disjoint dim subsets) ----
        float pn_part = 0.f;
#pragma unroll
        for (int kt = 0; kt < KT; ++kt)
            pn_part += pa[kt].x * pa[kt].x + pa[kt].y * pa[kt].y;
        const float pnorm2 = pn_part + __shfl_xor(pn_part, 16, 32); // for point s

        // ---- GEMM1: dots = P x At^T  (C layout: m = point, n = attractor) ----
        v8f c1 = {0.f, 0.f, 0.f, 0.f, 0.f, 0.f, 0.f, 0.f};
#pragma unroll
        for (int kt = 0; kt < KT; ++kt)
            c1 = __builtin_amdgcn_wmma_f32_16x16x4_f32(
                     false, pa[kt], false, b1[kt], (short)0, c1, false, false);

        // ---- w[s,a] = G / (sqrt(d2)+eps)^3 ; stash W tile in LDS ----
#pragma unroll
        for (int r = 0; r < 8; ++r) {
            const int sp = r + 8 * h;                  // point index of C element
            const float pn2  = __shfl(pnorm2, sp, 32); // |p_sp|^2 (replicated)
            const float d2   = fmaxf(anorm2 + pn2 - 2.f * c1[r], 0.f);
            const float dist = sqrtf(d2) + EPS;
            const float w    = amask * GRAVITY / (dist * dist * dist);
            W_lds[wave][sp][s] = w;
        }

        // ---- reload W as B-fragments (W^T: k = attractor, n = point) ----
        v2f b2[4];
        float ws_part = 0.f;
#pragma unroll
        for (int kt = 0; kt < 4; ++kt) {
            b2[kt] = *(const v2f*)&W_lds[wave][s][4 * kt + 2 * h];
            ws_part += b2[kt].x + b2[kt].y;
        }
        const float wsum = ws_part + __shfl_xor(ws_part, 16, 32); // sum_a w[s,a]

        // ---- GEMM2 (transposed): F^T tiles = At^T x W^T ; spill to LDS ----
#pragma unroll
        for (int dt = 0; dt < DT; ++dt) {
            v8f c2 = {0.f, 0.f, 0.f, 0.f, 0.f, 0.f, 0.f, 0.f};
#pragma unroll
            for (int kt = 0; kt < 4; ++kt)
                c2 = __builtin_amdgcn_wmma_f32_16x16x4_f32(
                         false, a2[dt][kt], false, b2[kt], (short)0, c2, false, false);
#pragma unroll
            for (int r = 0; r < 8; ++r)
                F_lds[wave][s][dt * 16 + r + 8 * h] = c2[r];
        }

        // ---- p += STEP * ( (W@At)[s,d] - p * wsum ) ----
#pragma unroll
        for (int kt = 0; kt < KT; ++kt) {
            const v2f f = *(const v2f*)&F_lds[wave][s][4 * kt + 2 * h];
            pa[kt].x += STEP * (f.x - pa[kt].x * wsum);
            pa[kt].y += STEP * (f.y - pa[kt].y * wsum);
        }
    }

    // ---- write back final positions ----
#pragma unroll
    for (int kt = 0; kt < KT; ++kt)
        *(v2f*)(out + (base + s) * DIM + 4 * kt + 2 * h) = pa[kt];
}

extern "C" void kernel_launch(void* const* d_in, const int* in_sizes, int n_in,
                              void* d_out, int out_size, void* d_ws, size_t ws_size,
                              hipStream_t stream) {
    const float* emb    = (const float*)d_in[0];  // [4,4096,128] f32
    const float* attr   = (const float*)d_in[1];  // [10,128]     f32
    const int*   n_iter = (const int*)d_in[2];    // scalar int
    float*       out    = (float*)d_out;          // [4,4096,128] f32

    const int n_points = in_sizes[0] / DIM;       // 16384
    const int tiles    = n_points / 16;           // 1024 wave-tiles
    const int blocks   = tiles / 4;               // 4 waves (128 thr) per block

    attractor_wmma_kernel<<<blocks, 128, 0, stream>>>(emb, attr, n_iter, out);
}